// PointCloudSO3RotRandomizer_20461224198588
// MI455X (gfx1250) — compile-verified
//
#include <hip/hip_runtime.h>
#include <stdint.h>
#include <stddef.h>

// ---------------- problem constants (from setup_inputs) ----------------
#define KK   99
#define BB   256
#define TT   2
#define NN   8192
#define TA   16
#define DROPP (1.0f/64.0f)
#define PI_F 3.14159265358979323846f

// output layout (flat, concatenated in return order)
#define OFF_PC   ((size_t)0)                       // 256*2*8192*6 = 25165824
#define OFF_POS  ((size_t)25165824)                // +1536
#define OFF_QUAT ((size_t)25167360)                // +2048
#define OFF_NACT ((size_t)25169408)                // +40960 -> 25210368 total

typedef __attribute__((ext_vector_type(2))) float v2f;
typedef __attribute__((ext_vector_type(8))) float v8f;

// ---------------- CDNA5 async-copy helpers (inline asm) ----------------
__device__ __forceinline__ void wait_async0() {
  asm volatile("s_wait_asynccnt 0" ::: "memory");
}
__device__ __forceinline__ void wait_ds0() {
  asm volatile("s_wait_dscnt 0" ::: "memory");
}
// copy 1536 contiguous bytes (64 points * 24B): 3 x b128 per lane
__device__ __forceinline__ void async_load3(unsigned lds_off, const float* gsrc) {
  asm volatile(
      "global_load_async_to_lds_b128 %0, %1, off\n\t"
      "global_load_async_to_lds_b128 %0, %1, off offset:512\n\t"
      "global_load_async_to_lds_b128 %0, %1, off offset:1024"
      :: "v"(lds_off), "v"(gsrc) : "memory");
}
__device__ __forceinline__ void async_store3(float* gdst, unsigned lds_off) {
  asm volatile(
      "global_store_async_from_lds_b128 %0, %1, off\n\t"
      "global_store_async_from_lds_b128 %0, %1, off offset:512\n\t"
      "global_store_async_from_lds_b128 %0, %1, off offset:1024"
      :: "v"(gdst), "v"(lds_off) : "memory");
}

// theta for (k,b): ang = where(drop<p, 0, u-0.5) * pi   (only Z survives AUG)
__device__ __forceinline__ float theta_of(const float* __restrict__ ang_u,
                                          const float* __restrict__ drop_u,
                                          int k, int b) {
  float u  = ang_u[((size_t)k * BB + b) * 3 + 2];
  float du = drop_u[(size_t)k * BB + b];
  float a  = (du < DROPP) ? 0.0f : (u - 0.5f);
  return a * PI_F;
}

// ---------------- kernel 1: per-k validity (ok[k]) ----------------
__global__ void k_check(const float* __restrict__ pos,
                        const float* __restrict__ nact,
                        const float* __restrict__ ang_u,
                        const float* __restrict__ drop_u,
                        int* __restrict__ ok) {
  const int k = blockIdx.x;      // 0..98
  const int b = threadIdx.x;     // 0..255
  float th = theta_of(ang_u, drop_u, k, b);
  float s = sinf(th), c = cosf(th);
  bool good = true;
  #pragma unroll
  for (int t = 0; t < TT; ++t) {
    const float* p = pos + ((size_t)b * TT + t) * 3;
    float x = p[0], y = p[1], z = p[2];
    float xp = c * x - s * y;
    float yp = s * x + c * y;
    good = good && (fabsf(xp) <= 1.0f) && (fabsf(yp) <= 1.0f) && (fabsf(z) <= 1.0f);
  }
  #pragma unroll 4
  for (int ta = 0; ta < TA; ++ta) {
    const float* a = nact + ((size_t)b * TA + ta) * 10;
    float x = a[0], y = a[1];
    float xp = c * x - s * y;
    float yp = s * x + c * y;
    good = good && (fabsf(xp) <= 1.0f) && (fabsf(yp) <= 1.0f);
  }
  __shared__ int flag;
  if (threadIdx.x == 0) flag = 1;
  __syncthreads();
  if (!good) atomicAnd(&flag, 0);
  __syncthreads();
  if (threadIdx.x == 0) ok[k] = flag;
}

// ---------------- kernel 2: pick idx, build R[b], small outputs ----------------
__global__ void k_select(const float* __restrict__ pos,
                         const float* __restrict__ quat,
                         const float* __restrict__ nact,
                         const float* __restrict__ ang_u,
                         const float* __restrict__ drop_u,
                         const int* __restrict__ ok,
                         float* __restrict__ Rws,
                         float* __restrict__ out) {
  __shared__ int sidx;
  if (threadIdx.x == 0) sidx = KK - 1;
  __syncthreads();
  int t = threadIdx.x;
  if (t < KK && ok[t]) atomicMin(&sidx, t);
  __syncthreads();
  const int idx = sidx;
  const int b = threadIdx.x;     // one thread per batch element

  float th = theta_of(ang_u, drop_u, idx, b);
  float s = sinf(th), c = cosf(th);
  float R[9] = { c, -s, 0.f,
                 s,  c, 0.f,
                 0.f, 0.f, 1.f };
  #pragma unroll
  for (int i = 0; i < 9; ++i) Rws[b * 9 + i] = R[i];

  float* opos  = out + OFF_POS;
  float* oquat = out + OFF_QUAT;
  float* onact = out + OFF_NACT;

  // rotated_pos & rotated_quat
  #pragma unroll
  for (int tt = 0; tt < TT; ++tt) {
    const float* p = pos + ((size_t)b * TT + tt) * 3;
    float x = p[0], y = p[1], z = p[2];
    float* po = opos + ((size_t)b * TT + tt) * 3;
    po[0] = R[0]*x + R[1]*y + R[2]*z;
    po[1] = R[3]*x + R[4]*y + R[5]*z;
    po[2] = R[6]*x + R[7]*y + R[8]*z;

    // quat input is xyzw; reference reorders to wxyz
    const float* q = quat + ((size_t)b * TT + tt) * 4;
    float qr = q[3], qi = q[0], qj = q[1], qk = q[2];
    float two_s = 2.0f / (qr*qr + qi*qi + qj*qj + qk*qk);
    float m[9];
    m[0] = 1.f - two_s * (qj*qj + qk*qk);
    m[1] = two_s * (qi*qj - qk*qr);
    m[2] = two_s * (qi*qk + qj*qr);
    m[3] = two_s * (qi*qj + qk*qr);
    m[4] = 1.f - two_s * (qi*qi + qk*qk);
    m[5] = two_s * (qj*qk - qi*qr);
    m[6] = two_s * (qi*qk - qj*qr);
    m[7] = two_s * (qj*qk + qi*qr);
    m[8] = 1.f - two_s * (qi*qi + qj*qj);
    float n[9];
    #pragma unroll
    for (int i = 0; i < 3; ++i)
      #pragma unroll
      for (int kcol = 0; kcol < 3; ++kcol)
        n[i*3+kcol] = R[i*3+0]*m[0*3+kcol] + R[i*3+1]*m[1*3+kcol] + R[i*3+2]*m[2*3+kcol];

    // matrix_to_quaternion (wxyz), then reorder to xyzw
    float m00=n[0], m01=n[1], m02=n[2], m10=n[3], m11=n[4], m12=n[5], m20=n[6], m21=n[7], m22=n[8];
    float qa0 = sqrtf(fmaxf(1.f + m00 + m11 + m22, 0.f));
    float qa1 = sqrtf(fmaxf(1.f + m00 - m11 - m22, 0.f));
    float qa2 = sqrtf(fmaxf(1.f - m00 + m11 - m22, 0.f));
    float qa3 = sqrtf(fmaxf(1.f - m00 - m11 + m22, 0.f));
    int bi = 0; float bv = qa0;
    if (qa1 > bv) { bv = qa1; bi = 1; }
    if (qa2 > bv) { bv = qa2; bi = 2; }
    if (qa3 > bv) { bv = qa3; bi = 3; }
    float cw, cx, cy, cz;
    if      (bi == 0) { cw = qa0*qa0;  cx = m21-m12; cy = m02-m20; cz = m10-m01; }
    else if (bi == 1) { cw = m21-m12;  cx = qa1*qa1; cy = m10+m01; cz = m02+m20; }
    else if (bi == 2) { cw = m02-m20;  cx = m10+m01; cy = qa2*qa2; cz = m12+m21; }
    else              { cw = m10-m01;  cx = m20+m02; cy = m21+m12; cz = qa3*qa3; }
    float inv_d = 1.0f / (2.0f * fmaxf(bv, 0.1f));
    float* qo = oquat + ((size_t)b * TT + tt) * 4;
    qo[0] = cx * inv_d; qo[1] = cy * inv_d; qo[2] = cz * inv_d; qo[3] = cw * inv_d;
  }

  // rotated_naction: [R@apos, first-two-rows of R@rot6d_to_matrix, grip]
  for (int ta = 0; ta < TA; ++ta) {
    const float* a = nact + ((size_t)b * TA + ta) * 10;
    float* o = onact + ((size_t)b * TA + ta) * 10;
    float x = a[0], y = a[1], z = a[2];
    o[0] = R[0]*x + R[1]*y + R[2]*z;
    o[1] = R[3]*x + R[4]*y + R[5]*z;
    o[2] = R[6]*x + R[7]*y + R[8]*z;

    float a1x = a[3], a1y = a[4], a1z = a[5];
    float a2x = a[6], a2y = a[7], a2z = a[8];
    float inv1 = 1.0f / sqrtf(a1x*a1x + a1y*a1y + a1z*a1z);
    float b1x = a1x*inv1, b1y = a1y*inv1, b1z = a1z*inv1;
    float dt = b1x*a2x + b1y*a2y + b1z*a2z;
    float ux = a2x - dt*b1x, uy = a2y - dt*b1y, uz = a2z - dt*b1z;
    float inv2 = 1.0f / sqrtf(ux*ux + uy*uy + uz*uz);
    float b2x = ux*inv2, b2y = uy*inv2, b2z = uz*inv2;
    float b3x = b1y*b2z - b1z*b2y;
    float b3y = b1z*b2x - b1x*b2z;
    float b3z = b1x*b2y - b1y*b2x;
    float ar[9] = { b1x, b1y, b1z,  b2x, b2y, b2z,  b3x, b3y, b3z };
    // first two rows of R @ ar
    #pragma unroll
    for (int i = 0; i < 2; ++i)
      #pragma unroll
      for (int kcol = 0; kcol < 3; ++kcol)
        o[3 + i*3 + kcol] = R[i*3+0]*ar[0*3+kcol] + R[i*3+1]*ar[1*3+kcol] + R[i*3+2]*ar[2*3+kcol];
    o[9] = a[9];
  }
}

// ---------------- kernel 3: point-cloud rotation (bandwidth kernel) ----------------
// grid = 1024 blocks * 256 threads. Each block covers 4096 points of one b
// (4 blocks per b). Each wave independently streams 8 chunks of 64 points
// (1536 B) through double-buffered LDS using CDNA5 async global<->LDS copies,
// rotating xyz with V_WMMA_F32_16X16X4_F32 (A = R padded 16x4, B = 16 points
// as 4x16 columns, D rows 0..2 = rotated xyz per column/lane).
__global__ void __launch_bounds__(256)
k_rotate_pc(const float* __restrict__ pc,
            const float* __restrict__ Rws,
            float* __restrict__ out) {
  __shared__ float tile[2][8][64 * 6];   // 2 bufs x 8 waves x 1536 B = 24 KB

  const int tid  = threadIdx.x;
  const int lane = tid & 31;
  const int wave = tid >> 5;
  const int n16  = lane & 15;
  const bool hi  = lane >= 16;

  const int b = blockIdx.x >> 2;          // 4096 pts/block, 16384 pts per b
  float R[9];
  #pragma unroll
  for (int i = 0; i < 9; ++i) R[i] = Rws[b * 9 + i];

  // A operand: 16x4 f32 layout -> VGPR0: lanes0-15 K=0 (M=lane), lanes16-31 K=2;
  //            VGPR1: lanes0-15 K=1, lanes16-31 K=3(=0). Rows >=3 are zero.
  float a0, a1;
  if (!hi) {
    a0 = (lane < 3) ? R[lane * 3 + 0] : 0.0f;
    a1 = (lane < 3) ? R[lane * 3 + 1] : 0.0f;
  } else {
    int m = lane - 16;
    a0 = (m < 3) ? R[m * 3 + 2] : 0.0f;
    a1 = 0.0f;
  }
  const v2f Amat = { a0, a1 };

  // 32-bit LDS byte offsets for the async-copy asm (low 32 bits of the
  // generic address of an LDS object are the LDS offset).
  unsigned ldsoff[2];
  ldsoff[0] = (unsigned)(uintptr_t)&tile[0][wave][0] + (unsigned)lane * 16u;
  ldsoff[1] = (unsigned)(uintptr_t)&tile[1][wave][0] + (unsigned)lane * 16u;

  const size_t blockBase = (size_t)blockIdx.x * 4096;

  for (int cIt = 0; cIt < 8; ++cIt) {
    const int buf = cIt & 1;
    const size_t cp = blockBase + (size_t)(cIt * 8 + wave) * 64;  // chunk point base
    const float* gs = pc + cp * 6 + (size_t)lane * 4;
    async_load3(ldsoff[buf], gs);          // overlaps with prev chunk's stores
    wait_async0();                         // loads landed (and prev stores done)

    #pragma unroll
    for (int g = 0; g < 64; g += 16) {
      // B operand 4x16: VGPR0 lanes0-15 = row0 (x_n), lanes16-31 = row2 (z_n);
      //                 VGPR1 lanes0-15 = row1 (y_n), lanes16-31 = row3 (0).
      // Direct addrspace(3) indexing so these lower to ds_load/ds_store.
      float bx = tile[buf][wave][(g + n16) * 6 + (hi ? 2 : 0)];
      float by = tile[buf][wave][(g + n16) * 6 + 1];
      v2f Bmat = { bx, hi ? 0.0f : by };
      v8f Cz = { 0.f, 0.f, 0.f, 0.f, 0.f, 0.f, 0.f, 0.f };
      v8f D = __builtin_amdgcn_wmma_f32_16x16x4_f32(
          /*neg_a=*/false, Amat, /*neg_b=*/false, Bmat,
          /*c_mod=*/(short)0, Cz, /*reuse_a=*/false, /*reuse_b=*/false);
      // D: VGPR r holds (M=r, N=lane) for lanes 0-15 -> rotated xyz of point g+lane
      if (!hi) {
        tile[buf][wave][(g + lane) * 6 + 0] = D[0];
        tile[buf][wave][(g + lane) * 6 + 1] = D[1];
        tile[buf][wave][(g + lane) * 6 + 2] = D[2];
      }
    }
    wait_ds0();                            // LDS writes complete before async engine reads
    float* gd = out + OFF_PC + cp * 6 + (size_t)lane * 4;
    async_store3(gd, ldsoff[buf]);         // left in flight; next iter overlaps
  }
  wait_async0();
}

// ---------------- launcher ----------------
extern "C" void kernel_launch(void* const* d_in, const int* in_sizes, int n_in,
                              void* d_out, int out_size, void* d_ws, size_t ws_size,
                              hipStream_t stream) {
  const float* pc     = (const float*)d_in[0];
  const float* pos    = (const float*)d_in[1];
  const float* quat   = (const float*)d_in[2];
  const float* nact   = (const float*)d_in[3];
  const float* ang_u  = (const float*)d_in[4];
  const float* drop_u = (const float*)d_in[5];
  float* out = (float*)d_out;

  int*   ok  = (int*)d_ws;                       // 99 ints
  float* Rws = (float*)((char*)d_ws + 512);      // 256*9 floats

  k_check<<<KK, BB, 0, stream>>>(pos, nact, ang_u, drop_u, ok);
  k_select<<<1, BB, 0, stream>>>(pos, quat, nact, ang_u, drop_u, ok, Rws, out);
  k_rotate_pc<<<1024, 256, 0, stream>>>(pc, Rws, out);
}